// PytorchMetricLearningObjectiveWithSampling_24137716204076
// MI455X (gfx1250) — compile-verified
//
#include <hip/hip_runtime.h>
#include <hip/hip_bf16.h>
#include <math.h>

typedef __attribute__((ext_vector_type(2))) float v2f;
typedef __attribute__((ext_vector_type(8))) float v8f;

#define BS        128
#define SEGM      32
#define DIM       128
#define HALFB     64
#define NPOS      192          // 64*3
#define NE        384          // 192 pos + 192 neg
#define NX        1984         // 64*31
#define TILES_E   24           // 384/16
#define XM        64           // smooth macro-tile edge
#define XBLK      (NX / XM)    // 31
#define LDSPITCH  132          // 128 + 4 pad -> bank rotation of 4/row
#define MARGIN_F  0.05f
#define MSEPS_F   0.1f
#define LAMBDA_F  8e-05f

// ---------------------------------------------------------------- init
__global__ void k_init(float* acc) {
    acc[0] = 0.f;                     // loss_sum
    ((unsigned int*)acc)[1] = 0u;     // nnz
    acc[2] = 0.f;                     // smooth_sum
}

// ---------------------------------------------------------------- selection
__device__ inline void top3_of32(const float* d, int* out) {
    unsigned used = 0;
    for (int k = 0; k < 3; ++k) {
        int best = 0; float bv = -1e30f;
        for (int s = 0; s < SEGM; ++s) {
            if (!((used >> s) & 1u) && d[s] > bv) { bv = d[s]; best = s; }
        }
        used |= (1u << best);
        out[k] = best;
    }
}

// one block per anomal row (64 blocks, 128 threads)
__global__ void k_select(const float* __restrict__ emb, const int* __restrict__ y,
                         float* __restrict__ e, float* __restrict__ en2,
                         float* __restrict__ x1, float* __restrict__ x2,
                         float* __restrict__ x1n2, float* __restrict__ x2n2)
{
    const int b = blockIdx.x;      // 0..63
    const int t = threadIdx.x;     // 0..127
    __shared__ float anchor[DIM];
    __shared__ float red[DIM];
    __shared__ float d_an[SEGM], d_no[SEGM];
    __shared__ int   sel_p[3], sel_n[3];
    __shared__ int   s_an, s_no;

    if (t == 0) {  // stable partition: b-th label-1 index, b-th label-0 index
        int ca = 0, cn = 0, ai = 0, ni = 0;
        for (int q = 0; q < BS; ++q) {
            if (y[q] != 0) { if (ca == b) ai = q; ++ca; }
            else           { if (cn == b) ni = q; ++cn; }
        }
        s_an = ai; s_no = ni;
    }
    __syncthreads();
    const int an = s_an, no = s_no;

    anchor[t] = emb[(an * SEGM + 0) * DIM + t];
    __syncthreads();

    if (t < SEGM) {
        const float* row = emb + (an * SEGM + t) * DIM;
        float s = 0.f;
        for (int c = 0; c < DIM; ++c) { float d = row[c] - anchor[c]; s += d * d; }
        d_an[t] = s;
    } else if (t < 2 * SEGM) {
        const int sg = t - SEGM;
        const float* row = emb + (no * SEGM + sg) * DIM;
        float s = 0.f;
        for (int c = 0; c < DIM; ++c) { float d = row[c] - anchor[c]; s += d * d; }
        d_no[sg] = s;
    }
    __syncthreads();
    if (t == 0) top3_of32(d_an, sel_p);   // sqrt is monotone: same top-k as d_norm
    if (t == 1) top3_of32(d_no, sel_n);
    __syncthreads();

    // write 6 selected rows of e (l2-normalized) + their squared norms
    for (int r = 0; r < 6; ++r) {
        const float* src; int dst;
        if (r < 3) { src = emb + (an * SEGM + sel_p[r]) * DIM;     dst = b * 3 + r; }
        else       { src = emb + (no * SEGM + sel_n[r - 3]) * DIM; dst = NPOS + b * 3 + (r - 3); }
        float v = src[t];
        red[t] = v * v; __syncthreads();
        for (int off = 64; off > 0; off >>= 1) { if (t < off) red[t] += red[t + off]; __syncthreads(); }
        float S = red[0]; __syncthreads();
        float den = sqrtf(S + 1e-12f);
        e[dst * DIM + t] = v / den;
        if (t == 0) en2[dst] = S / (den * den);
    }

    // smoothness operands: x1 = segments 1..31, x2 = segments 0..30 (normalized)
    for (int s = 0; s < SEGM - 1; ++s) {
        for (int which = 0; which < 2; ++which) {
            const int seg = (which == 0) ? (s + 1) : s;
            const float* src = emb + (an * SEGM + seg) * DIM;
            float v = src[t];
            red[t] = v * v; __syncthreads();
            for (int off = 64; off > 0; off >>= 1) { if (t < off) red[t] += red[t + off]; __syncthreads(); }
            float S = red[0]; __syncthreads();
            float den = sqrtf(S + 1e-12f);
            const int row = b * (SEGM - 1) + s;
            float* dp = (which == 0) ? x1 : x2;
            float* np = (which == 0) ? x1n2 : x2n2;
            dp[row * DIM + t] = v / den;
            if (t == 0) np[row] = S / (den * den);
        }
    }
}

// ---------------------------------------------------------------- sim = e @ e^T  (fp32 WMMA, global-fed; L2 resident)
__global__ void k_sim_wmma(const float* __restrict__ e, float* __restrict__ sim)
{
    const int lane = threadIdx.x & 31;
    const int wave = threadIdx.x >> 5;
    const int tile = blockIdx.x * 4 + wave;        // 576 tiles exactly
    const int tm = tile / TILES_E, tn = tile % TILES_E;
    const int hi = lane >> 4;        // K sub-pair select
    const int l15 = lane & 15;

    const float* ar = e + (tm * 16 + l15) * DIM + hi * 2;
    const float* br = e + (tn * 16 + l15) * DIM + hi * 2;

    v8f c = {};
    #pragma unroll 4
    for (int k0 = 0; k0 < DIM; k0 += 4) {
        v2f a = *(const v2f*)(ar + k0);
        v2f b = *(const v2f*)(br + k0);
        c = __builtin_amdgcn_wmma_f32_16x16x4_f32(false, a, false, b, (short)0, c, false, false);
    }
    const int col = tn * 16 + l15;
    #pragma unroll
    for (int v = 0; v < 8; ++v) {
        const int row = tm * 16 + v + hi * 8;
        sim[row * NE + col] = c[v];
    }
}

// ---------------------------------------------------------------- per-row max_neg / min_pos
__global__ void k_rowstats(const float* __restrict__ sim,
                           float* __restrict__ maxneg, float* __restrict__ minpos)
{
    const int i = blockIdx.x;        // 0..383
    const int t = threadIdx.x;       // 128
    __shared__ float smx[128], smn[128];
    const bool ilab = (i < NPOS);
    float mx = -1e9f, mn = 1e9f;
    for (int j = t; j < NE; j += 128) {
        float s = sim[i * NE + j];
        const bool same = ((j < NPOS) == ilab);
        if (!same)          mx = fmaxf(mx, s);
        else if (j != i)    mn = fminf(mn, s);
    }
    smx[t] = mx; smn[t] = mn; __syncthreads();
    for (int off = 64; off > 0; off >>= 1) {
        if (t < off) { smx[t] = fmaxf(smx[t], smx[t + off]); smn[t] = fminf(smn[t], smn[t + off]); }
        __syncthreads();
    }
    if (t == 0) { maxneg[i] = smx[0]; minpos[i] = smn[0]; }
}

// ---------------------------------------------------------------- triplet accumulation
__global__ void k_triplet(const float* __restrict__ sim, const float* __restrict__ en2,
                          const float* __restrict__ maxneg, const float* __restrict__ minpos,
                          float* __restrict__ loss_sum, unsigned int* __restrict__ nnz)
{
    const int i = blockIdx.x;        // 0..383
    const int t = threadIdx.x;       // 256
    __shared__ float drow[NE];
    __shared__ unsigned char pm[NE], nm[NE];
    __shared__ float ssum[256];
    __shared__ unsigned int scnt[256];

    const bool ilab = (i < NPOS);
    const float mn = maxneg[i], mp = minpos[i], ni = en2[i];

    for (int j = t; j < NE; j += 256) {
        float s = sim[i * NE + j];
        float d2 = fmaxf(ni + en2[j] - 2.f * s, 0.f);
        drow[j] = (d2 > 0.f) ? sqrtf(d2) : 0.f;
        const bool jlab = (j < NPOS);
        pm[j] = (jlab == ilab) && (j != i) && (s - MSEPS_F < mn);
        nm[j] = (jlab != ilab) && (s + MSEPS_F > mp);
    }
    __syncthreads();

    float lsum = 0.f; unsigned int lcnt = 0;
    for (int j = 0; j < NE; ++j) {
        if (!pm[j]) continue;                 // uniform (shared) branch
        const float dj = drow[j] + MARGIN_F;
        for (int k = t; k < NE; k += 256) {
            if (nm[k]) {
                float viol = dj - drow[k];
                if (viol > 0.f) { lsum += viol; ++lcnt; }
            }
        }
    }
    ssum[t] = lsum; scnt[t] = lcnt; __syncthreads();
    for (int off = 128; off > 0; off >>= 1) {
        if (t < off) { ssum[t] += ssum[t + off]; scnt[t] += scnt[t + off]; }
        __syncthreads();
    }
    if (t == 0) { atomicAdd(loss_sum, ssum[0]); atomicAdd(nnz, scnt[0]); }
}

// ---------------------------------------------------------------- fused smooth:
// async-stage 64x64 macro tile operands into LDS, WMMA from LDS, fused sqrt+sum
__global__ void k_smooth_wmma(const float* __restrict__ x1, const float* __restrict__ x2,
                              const float* __restrict__ x1n2, const float* __restrict__ x2n2,
                              float* __restrict__ smooth_sum)
{
    __shared__ __align__(16) float sA[XM * LDSPITCH];
    __shared__ __align__(16) float sB[XM * LDSPITCH];

    const int t    = threadIdx.x;    // 256 = 8 waves
    const int lane = t & 31;
    const int wave = t >> 5;
    const int bm = blockIdx.x / XBLK;
    const int bn = blockIdx.x % XBLK;

    // ---- async stage: 64 rows x 128 cols of x1 and of x2, b128 chunks ----
    // 2048 chunks per matrix, 256 threads -> 8 chunks each per matrix.
    #pragma unroll
    for (int i = 0; i < 8; ++i) {
        const int c   = t + i * 256;     // 0..2047
        const int row = c >> 5;          // 0..63
        const int ch  = c & 31;          // 16B chunk within row
        {
            unsigned lds = (unsigned)(unsigned long long)&sA[row * LDSPITCH + ch * 4];
            unsigned long long ga =
                (unsigned long long)(x1 + ((size_t)(bm * XM + row) * DIM + ch * 4));
            asm volatile("global_load_async_to_lds_b128 %0, %1, off"
                         :: "v"(lds), "v"(ga) : "memory");
        }
        {
            unsigned lds = (unsigned)(unsigned long long)&sB[row * LDSPITCH + ch * 4];
            unsigned long long ga =
                (unsigned long long)(x2 + ((size_t)(bn * XM + row) * DIM + ch * 4));
            asm volatile("global_load_async_to_lds_b128 %0, %1, off"
                         :: "v"(lds), "v"(ga) : "memory");
        }
    }
    asm volatile("s_wait_asynccnt 0x0" ::: "memory");
    __syncthreads();

    // ---- 8 waves cover the 4x4 grid of 16x16 tiles; 2 tiles per wave ----
    const int hi  = lane >> 4;
    const int l15 = lane & 15;
    const int tc  = wave & 3;            // tile col 0..3
    float acc = 0.f;

    for (int tt = 0; tt < 2; ++tt) {
        const int tr = (wave >> 2) + tt * 2;   // tile row 0..3
        const float* ap = &sA[(tr * 16 + l15) * LDSPITCH + hi * 2];
        const float* bp = &sB[(tc * 16 + l15) * LDSPITCH + hi * 2];
        v8f c = {};
        #pragma unroll 4
        for (int k0 = 0; k0 < DIM; k0 += 4) {
            v2f a = *(const v2f*)(ap + k0);    // ds_load_b64, conflict-free (pitch 132)
            v2f b = *(const v2f*)(bp + k0);
            c = __builtin_amdgcn_wmma_f32_16x16x4_f32(false, a, false, b, (short)0, c, false, false);
        }
        const int col = bn * XM + tc * 16 + l15;
        const float cn2 = x2n2[col];
        #pragma unroll
        for (int v = 0; v < 8; ++v) {
            const int row = bm * XM + tr * 16 + v + hi * 8;
            float d2 = fmaxf(x1n2[row] + cn2 - 2.f * c[v], 0.f);
            acc += (d2 > 0.f) ? sqrtf(d2) : 0.f;
        }
    }
    for (int off = 16; off > 0; off >>= 1) acc += __shfl_down(acc, off, 32);
    if (lane == 0) atomicAdd(smooth_sum, acc);
}

// ---------------------------------------------------------------- finalize
__global__ void k_final(const float* __restrict__ acc, float* __restrict__ out)
{
    const float loss = acc[0];
    const float cnt = (float)((const unsigned int*)acc)[1];
    const float tl = loss / fmaxf(cnt, 1.f);
    out[0] = tl + LAMBDA_F * (acc[2] / ((float)NX * (float)NX));
}

// ================================================================ host
extern "C" void kernel_launch(void* const* d_in, const int* in_sizes, int n_in,
                              void* d_out, int out_size, void* d_ws, size_t ws_size,
                              hipStream_t stream) {
    const float* emb = (const float*)d_in[0];   // (128,32,128) f32
    const int*   y   = (const int*)d_in[1];     // (128,) i32

    float* ws = (float*)d_ws;
    size_t off = 0;
    float* e      = ws + off; off += (size_t)NE * DIM;   // 49152
    float* en2    = ws + off; off += NE;                 // 384
    float* sim    = ws + off; off += (size_t)NE * NE;    // 147456
    float* maxneg = ws + off; off += NE;
    float* minpos = ws + off; off += NE;
    float* x1     = ws + off; off += (size_t)NX * DIM;   // 253952
    float* x2     = ws + off; off += (size_t)NX * DIM;
    float* x1n2   = ws + off; off += NX;
    float* x2n2   = ws + off; off += NX;
    float* acc    = ws + off; off += 4;                  // [loss, nnz(u32), smooth, pad]

    k_init<<<1, 1, 0, stream>>>(acc);
    k_select<<<HALFB, 128, 0, stream>>>(emb, y, e, en2, x1, x2, x1n2, x2n2);
    k_sim_wmma<<<(TILES_E * TILES_E) / 4, 128, 0, stream>>>(e, sim);
    k_rowstats<<<NE, 128, 0, stream>>>(sim, maxneg, minpos);
    k_triplet<<<NE, 256, 0, stream>>>(sim, en2, maxneg, minpos, acc, (unsigned int*)acc + 1);
    k_smooth_wmma<<<XBLK * XBLK, 256, 0, stream>>>(x1, x2, x1n2, x2n2, acc + 2);
    k_final<<<1, 1, 0, stream>>>(acc, (float*)d_out);
}